// CantorGlobalAttention_14903536517167
// MI455X (gfx1250) — compile-verified
//
#include <hip/hip_runtime.h>
#include <cstdint>

// Problem constants (fixed by the reference).
#define E_      16
#define NPROJ_  2
#define B_      64
#define P_      4096
#define W_      3
#define CHUNK   1024          // p-elements handled per workgroup
#define NPLANES 14            // 2 Q-proj + 6 K-proj + 6 V-proj slices
#define THREADS 256           // 8 waves (wave32)

// Memory-bound elementwise "Cantor attention":
//   out[b, e*P+p] = softmax_w( Q*Kn*gate/scale ) . Vn
// Each workgroup owns one (expert, batch, p-chunk) tile. All 14 source planes
// for the tile are staged into LDS via gfx1250 async global->LDS b128 copies
// (ASYNCcnt), then the elementwise softmax runs out of LDS.
__global__ __launch_bounds__(THREADS) void cantor_attn_kernel(
    const float* __restrict__ Qp,     // [E, NPROJ, B, P]
    const float* __restrict__ Kp,     // [E, NPROJ, B, P]
    const float* __restrict__ Vp,     // [E, NPROJ, B, P]
    const float* __restrict__ betas,  // [E, W]
    const float* __restrict__ temp,   // [1]
    const int*   __restrict__ routes, // [E, W]
    float*       __restrict__ out)    // [B, E*P]
{
    __shared__ float smem[NPLANES * CHUNK];   // 56 KB

    const int t   = threadIdx.x;
    const int blk = blockIdx.x;               // E*B*(P/CHUNK) = 4096 blocks
    const int c   = blk & 3;                  // chunk index within P
    const int b   = (blk >> 2) & (B_ - 1);
    const int e   = blk >> 8;

    // Uniform per-workgroup: neighbor ids + sigmoid gates.
    int   r[W_];
    float g[W_];
#pragma unroll
    for (int w = 0; w < W_; ++w) {
        r[w] = routes[e * W_ + w];
        const float bw = betas[e * W_ + w];
        g[w] = (r[w] != e) ? (1.0f / (1.0f + __expf(-bw))) : 1.0f;
    }
    const float invScale = 1.0f / (11.31370849898476f * fabsf(temp[0])); // 1/(sqrt(128)*|T|)

    // 14 global source plane pointers for this (e,b,chunk) tile.
    const size_t pchunk = (size_t)c * CHUNK;
    const float* gsrc[NPLANES];
#pragma unroll
    for (int j = 0; j < NPROJ_; ++j)
        gsrc[j] = Qp + (((size_t)e * NPROJ_ + j) * B_ + b) * P_ + pchunk;
#pragma unroll
    for (int w = 0; w < W_; ++w) {
#pragma unroll
        for (int j = 0; j < NPROJ_; ++j) {
            const size_t off = (((size_t)r[w] * NPROJ_ + j) * B_ + b) * P_ + pchunk;
            gsrc[2 + 2 * w + j] = Kp + off;
            gsrc[8 + 2 * w + j] = Vp + off;
        }
    }

    // Stage all 14 planes into LDS with async b128 copies (16 B/lane, coalesced).
    // Low 32 bits of the generic shared pointer == LDS byte offset (aperture form).
    const uint32_t ldsbase = (uint32_t)(uintptr_t)(void*)smem;
    const int i = t * 4;                       // float4 slot within the chunk
#pragma unroll
    for (int pl = 0; pl < NPLANES; ++pl) {
        const uint32_t laddr = ldsbase + (uint32_t)((pl * CHUNK + i) * sizeof(float));
        const uint64_t gaddr = (uint64_t)(uintptr_t)(gsrc[pl] + i);
        asm volatile("global_load_async_to_lds_b128 %0, %1, off"
                     :: "v"(laddr), "v"(gaddr) : "memory");
    }
    asm volatile("s_wait_asynccnt 0" ::: "memory");
    __syncthreads();

    // Average the two projections, score, 3-way softmax, weighted V sum.
    float q[4], kk[W_][4], vv[W_][4];
    {
        const float4 a = *(const float4*)&smem[0 * CHUNK + i];
        const float4 d = *(const float4*)&smem[1 * CHUNK + i];
        q[0] = 0.5f * (a.x + d.x); q[1] = 0.5f * (a.y + d.y);
        q[2] = 0.5f * (a.z + d.z); q[3] = 0.5f * (a.w + d.w);
    }
#pragma unroll
    for (int w = 0; w < W_; ++w) {
        const float4 k0 = *(const float4*)&smem[(2 + 2 * w) * CHUNK + i];
        const float4 k1 = *(const float4*)&smem[(3 + 2 * w) * CHUNK + i];
        const float4 v0 = *(const float4*)&smem[(8 + 2 * w) * CHUNK + i];
        const float4 v1 = *(const float4*)&smem[(9 + 2 * w) * CHUNK + i];
        kk[w][0] = 0.5f * (k0.x + k1.x); kk[w][1] = 0.5f * (k0.y + k1.y);
        kk[w][2] = 0.5f * (k0.z + k1.z); kk[w][3] = 0.5f * (k0.w + k1.w);
        vv[w][0] = 0.5f * (v0.x + v1.x); vv[w][1] = 0.5f * (v0.y + v1.y);
        vv[w][2] = 0.5f * (v0.z + v1.z); vv[w][3] = 0.5f * (v0.w + v1.w);
    }

    float o[4];
#pragma unroll
    for (int u = 0; u < 4; ++u) {
        const float s0 = q[u] * kk[0][u] * invScale * g[0];
        const float s1 = q[u] * kk[1][u] * invScale * g[1];
        const float s2 = q[u] * kk[2][u] * invScale * g[2];
        const float m  = fmaxf(s0, fmaxf(s1, s2));
        const float e0 = __expf(s0 - m);
        const float e1 = __expf(s1 - m);
        const float e2 = __expf(s2 - m);
        const float inv = 1.0f / (e0 + e1 + e2);
        o[u] = (e0 * vv[0][u] + e1 * vv[1][u] + e2 * vv[2][u]) * inv;
    }

    // out[b, e*P + p] : contiguous b128 store.
    *(float4*)&out[(((size_t)b * E_ + e) * P_) + pchunk + i] =
        make_float4(o[0], o[1], o[2], o[3]);
}

extern "C" void kernel_launch(void* const* d_in, const int* in_sizes, int n_in,
                              void* d_out, int out_size, void* d_ws, size_t ws_size,
                              hipStream_t stream) {
    const float* Qp     = (const float*)d_in[0];
    const float* Kp     = (const float*)d_in[1];
    const float* Vp     = (const float*)d_in[2];
    const float* betas  = (const float*)d_in[3];
    const float* temp   = (const float*)d_in[4];
    const int*   routes = (const int*)d_in[5];
    float*       out    = (float*)d_out;
    (void)in_sizes; (void)n_in; (void)out_size; (void)d_ws; (void)ws_size;

    const dim3 grid(E_ * B_ * (P_ / CHUNK));   // 4096 workgroups
    cantor_attn_kernel<<<grid, THREADS, 0, stream>>>(Qp, Kp, Vp, betas, temp, routes, out);
}